// LeNet5_T_7679401525820
// MI455X (gfx1250) — compile-verified
//
#include <hip/hip_runtime.h>
#include <hip/hip_bf16.h>
#include <math.h>

typedef __attribute__((ext_vector_type(2))) float v2f;
typedef __attribute__((ext_vector_type(8))) float v8f;

#define BN_INV 0.9999950000374997f   // 1/sqrt(1 + 1e-5)

__device__ __forceinline__ float wave_sum32(float v) {
#pragma unroll
  for (int off = 16; off >= 1; off >>= 1) v += __shfl_xor(v, off, 32);
  return v;
}

// ---------------------------------------------------------------- K0: W1^T
__global__ void transpose_w1(const float* __restrict__ w, float* __restrict__ wt) {
  int i = blockIdx.x * blockDim.x + threadIdx.x;   // 512*1024 elements
  if (i < 512 * 1024) {
    int k = i >> 9, n = i & 511;                   // wt[k][n] = w[n][k]
    wt[i] = w[n * 1024 + k];
  }
}

// ------------------------------------------------- K1: conv1+tern+bn+pool
__device__ __forceinline__ float conv1_at(const float* __restrict__ xs,
                                          const float* __restrict__ wc,
                                          float bias, int oy, int ox) {
  float acc = bias;
  const float* xp = xs + oy * 28 + ox;
#pragma unroll
  for (int ky = 0; ky < 5; ++ky)
#pragma unroll
    for (int kx = 0; kx < 5; ++kx)
      acc += xp[ky * 28 + kx] * wc[ky * 5 + kx];
  return acc;
}

__global__ __launch_bounds__(256)
void conv1_tern_pool(const float* __restrict__ x, const float* __restrict__ w,
                     const float* __restrict__ b, const float* __restrict__ g,
                     const float* __restrict__ bt, float* __restrict__ h1) {
  __shared__ float xs[784];
  __shared__ float ws_[800];
  __shared__ float bs[32];
  __shared__ float reda[8];
  __shared__ float redb[8];
  __shared__ float redc[8];

  const int tid = threadIdx.x;
  const int s = blockIdx.x;
  const float* xin = x + s * 784;
  for (int i = tid; i < 784; i += 256) xs[i] = xin[i];
  for (int i = tid; i < 800; i += 256) ws_[i] = w[i];
  if (tid < 32) bs[tid] = b[tid];
  __syncthreads();

  // pass A: sum|conv1| over all 32*24*24 outputs of this sample
  float asum = 0.f;
  for (int i = tid; i < 18432; i += 256) {
    int c = i / 576, p = i % 576, oy = p / 24, ox = p % 24;
    asum += fabsf(conv1_at(xs, ws_ + c * 25, bs[c], oy, ox));
  }
  asum = wave_sum32(asum);
  if ((tid & 31) == 0) reda[tid >> 5] = asum;
  __syncthreads();
  float tot = 0.f;
#pragma unroll
  for (int i = 0; i < 8; ++i) tot += reda[i];
  const float d = 0.7f * tot / 18432.0f;

  // pass B: strict-mask count + masked abs sum
  float cnt = 0.f, msum = 0.f;
  for (int i = tid; i < 18432; i += 256) {
    int c = i / 576, p = i % 576, oy = p / 24, ox = p % 24;
    float a = fabsf(conv1_at(xs, ws_ + c * 25, bs[c], oy, ox));
    if (a > d) { cnt += 1.f; msum += a; }
  }
  cnt = wave_sum32(cnt); msum = wave_sum32(msum);
  if ((tid & 31) == 0) { redb[tid >> 5] = cnt; redc[tid >> 5] = msum; }
  __syncthreads();
  float ct = 0.f, mt = 0.f;
#pragma unroll
  for (int i = 0; i < 8; ++i) { ct += redb[i]; mt += redc[i]; }
  const float alpha = mt / ct;

  // pass C: ternarize -> BN -> maxpool2 -> relu -> h1[s][32*12*12]
  for (int i = tid; i < 4608; i += 256) {
    int c = i / 144, p = i % 144, py = p / 12, px = p % 12;
    float gm = g[c] * BN_INV, bb = bt[c];
    float mx = -INFINITY;
#pragma unroll
    for (int dy = 0; dy < 2; ++dy)
#pragma unroll
      for (int dx = 0; dx < 2; ++dx) {
        float v = conv1_at(xs, ws_ + c * 25, bs[c], 2 * py + dy, 2 * px + dx);
        float t = (v > d) ? alpha : ((v < -d) ? -alpha : 0.f);
        mx = fmaxf(mx, t * gm + bb);
      }
    h1[s * 4608 + i] = fmaxf(mx, 0.f);
  }
}

// ----------------------------------- K2: conv2 (WMMA f32) + tern+bn+pool
__global__ __launch_bounds__(512)
void conv2_wmma(const float* __restrict__ h1, const float* __restrict__ w2,
                const float* __restrict__ b2, const float* __restrict__ g2,
                const float* __restrict__ bt2, float* __restrict__ h2) {
  __shared__ float h1s[4608];       // input sample [32][12][12]
  __shared__ float wch[64 * 80];    // weight K-chunk [oc][80]
  __shared__ float c2o[64 * 64];    // conv2 output [oc][pixel]
  __shared__ int   koff[800];       // im2col offset: k -> ic*144+ky*12+kx
  __shared__ float red[16];
  __shared__ float redb[16];

  const int tid = threadIdx.x;
  const int s = blockIdx.x;
  for (int i = tid; i < 4608; i += 512) h1s[i] = h1[s * 4608 + i];
  for (int k = tid; k < 800; k += 512) {
    int ic = k / 25, r = k % 25;
    koff[k] = ic * 144 + (r / 5) * 12 + (r % 5);
  }

  const int lane = tid & 31, wv = tid >> 5;
  const int lm = lane & 15, hi = lane >> 4;
  const int tm = wv >> 2, tn = wv & 3;       // 4x4 tiles of 16x16 over 64x64
  const int p = tn * 16 + lm;                // output pixel 0..63
  const int pb = (p >> 3) * 12 + (p & 7);    // base offset in h1s
  const int ocr = tm * 16 + lm;              // A-matrix row (out channel)

  v8f acc = {};
  for (int kc = 0; kc < 800; kc += 80) {
    __syncthreads();
    for (int i = tid; i < 64 * 80; i += 512) {
      int oc = i / 80, kk = i % 80;
      wch[i] = w2[oc * 800 + kc + kk];
    }
    __syncthreads();
#pragma unroll
    for (int kk = 0; kk < 80; kk += 4) {
      int kl = kk + hi * 2;                  // this lane's K pair within step
      v2f a, bf;
      a.x = wch[ocr * 80 + kl];
      a.y = wch[ocr * 80 + kl + 1];
      int kg = kc + kl;
      bf.x = h1s[koff[kg] + pb];
      bf.y = h1s[koff[kg + 1] + pb];
      acc = __builtin_amdgcn_wmma_f32_16x16x4_f32(
          false, a, false, bf, (short)0, acc, false, false);
    }
  }

  // C layout: VGPR r -> row (tm*16 + r + hi*8), col p ; add conv bias
#pragma unroll
  for (int r = 0; r < 8; ++r) {
    int oc = tm * 16 + r + hi * 8;
    c2o[oc * 64 + p] = acc[r] + b2[oc];
  }
  __syncthreads();

  // ternarize stats over all 4096 conv2 outputs
  float asum = 0.f;
  for (int i = tid; i < 4096; i += 512) asum += fabsf(c2o[i]);
  asum = wave_sum32(asum);
  if (lane == 0) red[wv] = asum;
  __syncthreads();
  float tot = 0.f;
#pragma unroll
  for (int i = 0; i < 16; ++i) tot += red[i];
  const float d = 0.7f * tot / 4096.f;
  __syncthreads();
  float cnt = 0.f, msum = 0.f;
  for (int i = tid; i < 4096; i += 512) {
    float a0 = fabsf(c2o[i]);
    if (a0 > d) { cnt += 1.f; msum += a0; }
  }
  cnt = wave_sum32(cnt); msum = wave_sum32(msum);
  if (lane == 0) { red[wv] = cnt; redb[wv] = msum; }
  __syncthreads();
  float ct = 0.f, mt = 0.f;
#pragma unroll
  for (int i = 0; i < 16; ++i) { ct += red[i]; mt += redb[i]; }
  const float alpha = mt / ct;

  // ternarize -> BN -> maxpool2 -> relu -> h2[s][1024]  ([64][4][4] flat)
  for (int i = tid; i < 1024; i += 512) {
    int c = i >> 4, pp = i & 15, py = pp >> 2, px = pp & 3;
    float gm = g2[c] * BN_INV, bb = bt2[c];
    float mx = -INFINITY;
#pragma unroll
    for (int dy = 0; dy < 2; ++dy)
#pragma unroll
      for (int dx = 0; dx < 2; ++dx) {
        float v = c2o[c * 64 + (2 * py + dy) * 8 + (2 * px + dx)];
        float t = (v > d) ? alpha : ((v < -d) ? -alpha : 0.f);
        mx = fmaxf(mx, t * gm + bb);
      }
    h2[s * 1024 + i] = fmaxf(mx, 0.f);
  }
}

// ---------------------- K3: fc1 (WMMA f32) + tern + relu + fused fc2+tern
__global__ __launch_bounds__(512)
void fc_wmma(const float* __restrict__ h2, const float* __restrict__ w1t,
             const float* __restrict__ b1, const float* __restrict__ w2f,
             const float* __restrict__ b2f, float* __restrict__ out) {
  __shared__ float o[16 * 512];              // fc1 outputs for 16 samples

  const int tid = threadIdx.x, lane = tid & 31, wv = tid >> 5;
  const int lm = lane & 15, hi = lane >> 4;
  const int rb = blockIdx.x * 16;            // first sample of this block
  const int n0 = (2 * wv) * 16 + lm;         // wave owns N-tiles 2w, 2w+1
  const int n1 = n0 + 16;
  const float* arow = h2 + (rb + lm) * 1024;

  v8f acc0 = {}, acc1 = {};
  for (int k0 = 0; k0 < 1024; k0 += 4) {
    int k = k0 + hi * 2;
    v2f a;  a.x = arow[k];               a.y = arow[k + 1];
    v2f bA; bA.x = w1t[k * 512 + n0];    bA.y = w1t[(k + 1) * 512 + n0];
    v2f bB; bB.x = w1t[k * 512 + n1];    bB.y = w1t[(k + 1) * 512 + n1];
    acc0 = __builtin_amdgcn_wmma_f32_16x16x4_f32(
        false, a, false, bA, (short)0, acc0, false, false);
    acc1 = __builtin_amdgcn_wmma_f32_16x16x4_f32(
        false, a, false, bB, (short)0, acc1, false, false);
  }
#pragma unroll
  for (int r = 0; r < 8; ++r) {
    int m = r + hi * 8;
    o[m * 512 + n0] = acc0[r] + b1[n0];
    o[m * 512 + n1] = acc1[r] + b1[n1];
  }
  __syncthreads();

  // wave wv owns sample row wv: ternarize(512) -> relu -> fc2 -> ternarize(10)
  float* row = o + wv * 512;
  float asum = 0.f;
  for (int i = lane; i < 512; i += 32) asum += fabsf(row[i]);
  asum = wave_sum32(asum);
  const float d = 0.7f * asum / 512.f;
  float cnt = 0.f, ms = 0.f;
  for (int i = lane; i < 512; i += 32) {
    float a0 = fabsf(row[i]);
    if (a0 > d) { cnt += 1.f; ms += a0; }
  }
  cnt = wave_sum32(cnt); ms = wave_sum32(ms);
  const float alpha = ms / cnt;
  for (int i = lane; i < 512; i += 32) {
    float v = row[i];
    float t = (v > d) ? alpha : ((v < -d) ? -alpha : 0.f);
    row[i] = fmaxf(t, 0.f);                  // relu(ternary)
  }

  // fc2: 10 outputs, K=512
  float acc10[10];
#pragma unroll
  for (int n2 = 0; n2 < 10; ++n2) {
    float a0 = 0.f;
    for (int k = lane; k < 512; k += 32) a0 += row[k] * w2f[n2 * 512 + k];
    acc10[n2] = wave_sum32(a0) + b2f[n2];    // all lanes hold the total
  }
  float s10 = 0.f;
#pragma unroll
  for (int n2 = 0; n2 < 10; ++n2) s10 += fabsf(acc10[n2]);
  const float d10 = 0.07f * s10;             // 0.7 * mean over n=10
  float c10 = 0.f, m10 = 0.f;
#pragma unroll
  for (int n2 = 0; n2 < 10; ++n2) {
    float a0 = fabsf(acc10[n2]);
    if (a0 > d10) { c10 += 1.f; m10 += a0; }
  }
  const float a10 = m10 / c10;
  if (lane == 0) {
    const int smp = rb + wv;
#pragma unroll
    for (int n2 = 0; n2 < 10; ++n2) {
      float v = acc10[n2];
      out[smp * 10 + n2] = ((v > d10) ? 1.f : ((v < -d10) ? -1.f : 0.f)) * a10;
    }
  }
}

// ---------------------------------------------------------------- launch
extern "C" void kernel_launch(void* const* d_in, const int* in_sizes, int n_in,
                              void* d_out, int out_size, void* d_ws, size_t ws_size,
                              hipStream_t stream) {
  const float* x   = (const float*)d_in[0];
  const float* c1w = (const float*)d_in[1];
  const float* c1b = (const float*)d_in[2];
  const float* g1  = (const float*)d_in[3];
  const float* bb1 = (const float*)d_in[4];
  const float* c2w = (const float*)d_in[5];
  const float* c2b = (const float*)d_in[6];
  const float* g2  = (const float*)d_in[7];
  const float* bb2 = (const float*)d_in[8];
  const float* f1w = (const float*)d_in[9];
  const float* f1b = (const float*)d_in[10];
  const float* f2w = (const float*)d_in[11];
  const float* f2b = (const float*)d_in[12];
  float* out = (float*)d_out;

  float* ws  = (float*)d_ws;
  float* w1t = ws;                            // 512*1024
  float* h1  = w1t + 512 * 1024;              // 2048*4608
  float* h2  = h1 + 2048 * 4608;              // 2048*1024

  transpose_w1   <<<2048, 256, 0, stream>>>(f1w, w1t);
  conv1_tern_pool<<<2048, 256, 0, stream>>>(x, c1w, c1b, g1, bb1, h1);
  conv2_wmma     <<<2048, 512, 0, stream>>>(h1, c2w, c2b, g2, bb2, h2);
  fc_wmma        <<<128,  512, 0, stream>>>(h2, w1t, f1b, f2w, f2b, out);
}